// SheafLaplacian_81698867905244
// MI455X (gfx1250) — compile-verified
//
#include <hip/hip_runtime.h>
#include <math.h>

// ---------------------------------------------------------------------------
// Sheaf Laplacian on MI455X (gfx1250, wave32).
// Output = dense 8192x8192 fp32 (256 MB, mostly zero) -> store-bandwidth bound
// (~11us floor @ 23.3 TB/s). Compute (~0.1 GFLOP of 8x8 matmuls) is mapped to
// V_WMMA_F32_16X16X4_F32 by packing two 8x8 problems block-diagonally into one
// 16x16 tile and accumulating K=16 as 4 chunks of K=4. Fragment builds are
// branch-free (unconditional in-bounds loads + cndmask zeroing) so no EXEC
// masking occurs around the WMMAs.
// ---------------------------------------------------------------------------

typedef float v2f __attribute__((ext_vector_type(2)));
typedef float v4f __attribute__((ext_vector_type(4)));
typedef float v8f __attribute__((ext_vector_type(8)));

__device__ __forceinline__ v8f wmma4(v2f a, v2f b, v8f c) {
  // (neg_a, A, neg_b, B, c_mod, C, reuse_a, reuse_b)
  return __builtin_amdgcn_wmma_f32_16x16x4_f32(false, a, false, b, (short)0, c,
                                               false, false);
}

// A fragment (16x4 slice kc of packed 16x16 = diag(A0, A1); A* 8x8 row-major).
// Lane L holds M = L%16; v_j = A[M, kbase+j], kbase = 4*kc + 2*(L>=16).
// Branch-free: load from the (always valid) owning matrix, zero via select.
__device__ __forceinline__ v2f fragA_pair(const float* __restrict__ A0,
                                          const float* __restrict__ A1,
                                          int kc, int lane, bool trans) {
  int i = lane & 15;
  int kbase = kc * 4 + ((lane >> 4) << 1);
  const float* base = (i < 8) ? A0 : A1;
  int ii = i & 7;
  v2f r;
#pragma unroll
  for (int j = 0; j < 2; ++j) {
    int kk = kbase + j;
    int k7 = kk & 7;
    float v = trans ? base[k7 * 8 + ii] : base[ii * 8 + k7];
    r[j] = ((i < 8) == (kk < 8)) ? v : 0.0f;  // block-diagonal mask
  }
  return r;
}

// B fragment (4x16 slice kc of packed diag(B0, B1)). Lane L holds N = L%16;
// v_j = B[kbase+j, N].
__device__ __forceinline__ v2f fragB_pair(const float* __restrict__ B0,
                                          const float* __restrict__ B1,
                                          int kc, int lane) {
  int n = lane & 15;
  int kbase = kc * 4 + ((lane >> 4) << 1);
  const float* base = (n < 8) ? B0 : B1;
  int n7 = n & 7;
  v2f r;
#pragma unroll
  for (int j = 0; j < 2; ++j) {
    int kk = kbase + j;
    float v = base[(kk & 7) * 8 + n7];
    r[j] = ((n < 8) == (kk < 8)) ? v : 0.0f;
  }
  return r;
}

// Fragments from a full 16x16 row-major LDS tile (off-blocks are exact zeros
// because both WMMA operands were block-diagonal).
__device__ __forceinline__ v2f fragA_lds(const float* __restrict__ t, int kc,
                                         int lane) {
  int i = lane & 15;
  int kbase = kc * 4 + ((lane >> 4) << 1);
  v2f r;
  r[0] = t[i * 16 + kbase + 0];
  r[1] = t[i * 16 + kbase + 1];
  return r;
}
__device__ __forceinline__ v2f fragB_lds(const float* __restrict__ t, int kc,
                                         int lane) {
  int n = lane & 15;
  int kbase = kc * 4 + ((lane >> 4) << 1);
  v2f r;
  r[0] = t[(kbase + 0) * 16 + n];
  r[1] = t[(kbase + 1) * 16 + n];
  return r;
}

// C layout (ISA 7.12.2): vgpr r, lane L -> (M = r + 8*(L>=16), N = L%16).
__device__ __forceinline__ void store_c_lds(float* __restrict__ t, v8f c,
                                            int lane) {
  int hi = (lane >> 4) & 1;
  int n = lane & 15;
#pragma unroll
  for (int r = 0; r < 8; ++r) t[(r + hi * 8) * 16 + n] = c[r];
}

// Extract this lane's element of its own 8x8 result block + target indices.
// Per vgpr r: lanes 0-7 own edge0 element (r, n); lanes 24-31 own edge1
// element (r, n-8); other lanes own packed zeros.
__device__ __forceinline__ bool c_owner(int lane, int* blk, int* col) {
  int hi = lane >> 4, n = lane & 15;
  *blk = hi;
  *col = n & 7;
  return (hi == 0 && n < 8) || (hi == 1 && n >= 8);
}

// ---------------------------------------------------------------------------
// Kernel 1: zero fill (nontemporal b128 stores) -- dominates runtime.
__global__ void k_zero4(v4f* __restrict__ p, size_t n4) {
  size_t i = (size_t)blockIdx.x * blockDim.x + threadIdx.x;
  size_t stride = (size_t)gridDim.x * blockDim.x;
  v4f z = {0.f, 0.f, 0.f, 0.f};
  for (; i < n4; i += stride) __builtin_nontemporal_store(z, p + i);
}

// ---------------------------------------------------------------------------
// Kernel 2: per-edge Gram R^T R via WMMA (2 edges/wave), atomic-add into
// diag[src].
__global__ void k_gram(const float* __restrict__ R, const int* __restrict__ src,
                       float* __restrict__ diagm, int npairs) {
  int lane = threadIdx.x & 31;
  int pair = blockIdx.x * 4 + (threadIdx.x >> 5);
  if (pair >= npairs) return;
  int e0 = 2 * pair, e1 = 2 * pair + 1;
  const float* Re0 = R + (size_t)e0 * 64;
  const float* Re1 = R + (size_t)e1 * 64;
  v8f c = {};
#pragma unroll
  for (int kc = 0; kc < 4; ++kc) {
    v2f a = fragA_pair(Re0, Re1, kc, lane, /*trans=*/true);  // A = R^T
    v2f b = fragB_pair(Re0, Re1, kc, lane);                  // B = R
    c = wmma4(a, b, c);
  }
  int s0 = src[e0], s1 = src[e1];
  int blk, col;
  bool own = c_owner(lane, &blk, &col);
  size_t base = (size_t)(blk ? s1 : s0) * 64 + col;
#pragma unroll
  for (int r = 0; r < 8; ++r)
    if (own) atomicAdd(&diagm[base + (size_t)r * 8], c[r]);
}

// ---------------------------------------------------------------------------
// Kernel 3: Dinv[v] = (diag[v] + 1e-3 I)^(-1/2), one node per thread via
// cyclic Jacobi on the 8x8 symmetric matrix (eigenvalues clipped at 1e-4).
__global__ void k_dinv(const float* __restrict__ diagm,
                       float* __restrict__ dinv, int n) {
  int v = blockIdx.x * blockDim.x + threadIdx.x;
  if (v >= n) return;
  float A[64], V[64];
  for (int i = 0; i < 8; ++i)
    for (int j = 0; j < 8; ++j) {
      A[i * 8 + j] = diagm[(size_t)v * 64 + i * 8 + j] + (i == j ? 1e-3f : 0.f);
      V[i * 8 + j] = (i == j) ? 1.f : 0.f;
    }
  for (int sweep = 0; sweep < 12; ++sweep) {
    for (int p = 0; p < 7; ++p)
      for (int q = p + 1; q < 8; ++q) {
        float apq = A[p * 8 + q];
        if (fabsf(apq) <= 1e-12f) continue;
        float app = A[p * 8 + p], aqq = A[q * 8 + q];
        float tau = (aqq - app) / (2.f * apq);
        float t =
            (tau >= 0.f ? 1.f : -1.f) / (fabsf(tau) + sqrtf(1.f + tau * tau));
        float cth = 1.f / sqrtf(1.f + t * t);
        float sth = t * cth;
        for (int k = 0; k < 8; ++k) {  // A = A * J  (cols p,q)
          float akp = A[k * 8 + p], akq = A[k * 8 + q];
          A[k * 8 + p] = cth * akp - sth * akq;
          A[k * 8 + q] = sth * akp + cth * akq;
        }
        for (int k = 0; k < 8; ++k) {  // A = J^T * A (rows p,q)
          float apk = A[p * 8 + k], aqk = A[q * 8 + k];
          A[p * 8 + k] = cth * apk - sth * aqk;
          A[q * 8 + k] = sth * apk + cth * aqk;
        }
        for (int k = 0; k < 8; ++k) {  // V = V * J
          float vkp = V[k * 8 + p], vkq = V[k * 8 + q];
          V[k * 8 + p] = cth * vkp - sth * vkq;
          V[k * 8 + q] = sth * vkp + cth * vkq;
        }
      }
  }
  float wi[8];
  for (int i = 0; i < 8; ++i) {
    float w = A[i * 8 + i];
    w = (w < 1e-4f) ? 1e-4f : w;
    wi[i] = 1.f / sqrtf(w);
  }
  for (int i = 0; i < 8; ++i)
    for (int j = 0; j < 8; ++j) {
      float acc = 0.f;
      for (int k = 0; k < 8; ++k) acc += V[i * 8 + k] * wi[k] * V[j * 8 + k];
      dinv[(size_t)v * 64 + i * 8 + j] = acc;
    }
}

// ---------------------------------------------------------------------------
// Kernel 4: diagonal blocks  out[v,v] = Dinv[v] * diag[v] * Dinv[v]
// (2 nodes per wave; chain through an LDS 16x16 tile).
__global__ void k_diagblk(const float* __restrict__ diagm,
                          const float* __restrict__ dinv,
                          float* __restrict__ out, int npairs, int ld) {
  __shared__ float tiles[4][256];
  int wid = threadIdx.x >> 5, lane = threadIdx.x & 31;
  int pair = blockIdx.x * 4 + wid;
  if (pair >= npairs) return;
  float* tile = tiles[wid];
  int v0 = 2 * pair, v1 = 2 * pair + 1;
  const float* D0 = dinv + (size_t)v0 * 64;
  const float* D1 = dinv + (size_t)v1 * 64;
  const float* G0 = diagm + (size_t)v0 * 64;
  const float* G1 = diagm + (size_t)v1 * 64;

  v8f c = {};
#pragma unroll
  for (int kc = 0; kc < 4; ++kc)  // M1 = Dinv * diag  (Dinv symmetric)
    c = wmma4(fragA_pair(D0, D1, kc, lane, false),
              fragB_pair(G0, G1, kc, lane), c);
  store_c_lds(tile, c, lane);
  __syncthreads();

  v8f c2 = {};
#pragma unroll
  for (int kc = 0; kc < 4; ++kc)  // M2 = M1 * Dinv
    c2 = wmma4(fragA_lds(tile, kc, lane), fragB_pair(D0, D1, kc, lane), c2);

  int blk, col;
  bool own = c_owner(lane, &blk, &col);
  int v = blk ? v1 : v0;
  size_t base = (size_t)(v * 8) * ld + v * 8 + col;
#pragma unroll
  for (int r = 0; r < 8; ++r)
    if (own) out[base + (size_t)r * ld] = c2[r];
}

// ---------------------------------------------------------------------------
// Kernel 5: off-diagonal blocks
//   out[src,dst] = -Dinv[src] * (R[rev]^T R[e]) * Dinv[dst]   (2 edges/wave)
__global__ void k_off(const float* __restrict__ R, const int* __restrict__ src,
                      const int* __restrict__ dst, const int* __restrict__ rev,
                      const float* __restrict__ dinv, float* __restrict__ out,
                      int npairs, int ld) {
  __shared__ float tiles[4][256];
  int wid = threadIdx.x >> 5, lane = threadIdx.x & 31;
  int pair = blockIdx.x * 4 + wid;
  if (pair >= npairs) return;
  float* tile = tiles[wid];
  int e0 = 2 * pair, e1 = 2 * pair + 1;
  int s0 = src[e0], s1 = src[e1];
  int t0 = dst[e0], t1 = dst[e1];
  const float* Re0 = R + (size_t)e0 * 64;
  const float* Re1 = R + (size_t)e1 * 64;
  const float* Rr0 = R + (size_t)rev[e0] * 64;
  const float* Rr1 = R + (size_t)rev[e1] * 64;

  v8f c = {};
#pragma unroll
  for (int kc = 0; kc < 4; ++kc)  // M1 = Rrev^T * Re
    c = wmma4(fragA_pair(Rr0, Rr1, kc, lane, /*trans=*/true),
              fragB_pair(Re0, Re1, kc, lane), c);
  store_c_lds(tile, c, lane);
  __syncthreads();

  const float* Ds0 = dinv + (size_t)s0 * 64;
  const float* Ds1 = dinv + (size_t)s1 * 64;
  v8f c2 = {};
#pragma unroll
  for (int kc = 0; kc < 4; ++kc)  // M2 = Dinv[s] * M1
    c2 = wmma4(fragA_pair(Ds0, Ds1, kc, lane, false),
               fragB_lds(tile, kc, lane), c2);
  __syncthreads();
  store_c_lds(tile, c2, lane);
  __syncthreads();

  const float* Dt0 = dinv + (size_t)t0 * 64;
  const float* Dt1 = dinv + (size_t)t1 * 64;
  v8f c3 = {};
#pragma unroll
  for (int kc = 0; kc < 4; ++kc)  // M3 = M2 * Dinv[t]
    c3 = wmma4(fragA_lds(tile, kc, lane), fragB_pair(Dt0, Dt1, kc, lane), c3);

  int blk, col;
  bool own = c_owner(lane, &blk, &col);
  int s = blk ? s1 : s0;
  int t = blk ? t1 : t0;
  size_t base = (size_t)(s * 8) * ld + t * 8 + col;
#pragma unroll
  for (int r = 0; r < 8; ++r)
    if (own) out[base + (size_t)r * ld] = -c3[r];
}

// ---------------------------------------------------------------------------
extern "C" void kernel_launch(void* const* d_in, const int* in_sizes, int n_in,
                              void* d_out, int out_size, void* d_ws,
                              size_t ws_size, hipStream_t stream) {
  const float* R = (const float*)d_in[0];
  const int* src = (const int*)d_in[1];
  const int* dst = (const int*)d_in[2];
  const int* rev = (const int*)d_in[3];
  float* out = (float*)d_out;

  const int E2 = in_sizes[1];                          // 2E directed edges
  const int nd = (int)(sqrt((double)out_size) + 0.5);  // n*d
  const int n = nd / 8;                                // nodes (d = 8)
  const int ld = nd;

  float* diagm = (float*)d_ws;           // n*64 floats
  float* dinv = diagm + (size_t)n * 64;  // n*64 floats
  (void)ws_size;
  (void)n_in;

  // 1) zero the 256 MB output (bandwidth-dominant step) + diag accumulator
  k_zero4<<<4096, 256, 0, stream>>>((v4f*)out, (size_t)out_size / 4);
  k_zero4<<<64, 256, 0, stream>>>((v4f*)diagm, (size_t)n * 64 / 4);

  // 2) Gram accumulation (WMMA, 2 edges/wave, 4 waves/block)
  int gpairs = E2 / 2;
  k_gram<<<(gpairs + 3) / 4, 128, 0, stream>>>(R, src, diagm, gpairs);

  // 3) per-node inverse sqrt of (diag + 1e-3 I)
  k_dinv<<<(n + 255) / 256, 256, 0, stream>>>(diagm, dinv, n);

  // 4) diagonal blocks
  int npairs = n / 2;
  k_diagblk<<<(npairs + 3) / 4, 128, 0, stream>>>(diagm, dinv, out, npairs, ld);

  // 5) off-diagonal blocks
  k_off<<<(gpairs + 3) / 4, 128, 0, stream>>>(R, src, dst, rev, dinv, out,
                                              gpairs, ld);
}